// PointNetGenerator_80410377716418
// MI455X (gfx1250) — compile-verified
//
#include <hip/hip_runtime.h>
#include <cstdint>

// ---------------------------------------------------------------------------
// PU-Net style PointNet++ generator for MI455X (gfx1250), fp32 WMMA path.
// All dense MLP math runs through v_wmma_f32_16x16x4_f32 (16x16 tiles, K=4),
// two accumulators per wave, guard-free inner loops.
// ---------------------------------------------------------------------------

#define BB        4
#define N0        4096
#define NSAMP     32
#define TPTS      2                  // sampled points per SA block
#define SROWS     (TPTS * NSAMP)     // 64 rows per SA block
#define SA_THREADS   256
#define GEMM_THREADS 256

typedef float v2f __attribute__((ext_vector_type(2)));
typedef float v8f __attribute__((ext_vector_type(8)));

__device__ __forceinline__ v8f wmma_f32_k4(v2f a, v2f b, v8f c) {
    // D = A(16x4,f32) x B(4x16,f32) + C(16x16,f32)
    return __builtin_amdgcn_wmma_f32_16x16x4_f32(
        /*neg_a=*/false, a, /*neg_b=*/false, b,
        /*c_mod=*/(short)0, c, /*reuse_a=*/false, /*reuse_b=*/false);
}

// ---------------------------------------------------------------------------
// Farthest point sampling: one block per batch. Matches jax scan semantics:
// first stored index is 0; argmax tie-break = smallest index.
// ---------------------------------------------------------------------------
__global__ void fps_kernel(const float* __restrict__ xyz, int N, int npoint,
                           int* __restrict__ idx) {
    __shared__ float dist[N0];
    __shared__ float rmax[256];
    __shared__ int   rarg[256];
    __shared__ int   curFar;
    const int b = blockIdx.x;
    const int tid = threadIdx.x;
    const float* px = xyz + (size_t)b * N * 3;
    for (int i = tid; i < N; i += blockDim.x) dist[i] = 1e10f;
    if (tid == 0) curFar = 0;
    __syncthreads();
    for (int it = 0; it < npoint; ++it) {
        const int far = curFar;
        if (tid == 0) idx[b * npoint + it] = far;
        const float cx = px[far * 3 + 0];
        const float cy = px[far * 3 + 1];
        const float cz = px[far * 3 + 2];
        float bm = -1.0f; int ba = 0;
        for (int i = tid; i < N; i += blockDim.x) {
            const float dx = px[i * 3 + 0] - cx;
            const float dy = px[i * 3 + 1] - cy;
            const float dz = px[i * 3 + 2] - cz;
            const float d  = dx * dx + dy * dy + dz * dz;
            const float nd = fminf(dist[i], d);
            dist[i] = nd;
            if (nd > bm) { bm = nd; ba = i; }
        }
        rmax[tid] = bm; rarg[tid] = ba;
        __syncthreads();
        for (int s = blockDim.x >> 1; s > 0; s >>= 1) {
            if (tid < s) {
                const float o = rmax[tid + s]; const int oa = rarg[tid + s];
                if (o > rmax[tid] || (o == rmax[tid] && oa < rarg[tid])) {
                    rmax[tid] = o; rarg[tid] = oa;
                }
            }
            __syncthreads();
        }
        if (tid == 0) curFar = rarg[0];
        __syncthreads();
    }
}

// new_xyz[b,j,:] = xyz[b, sidx[b,j], :]
__global__ void gather_xyz_kernel(const float* __restrict__ xyz,
                                  const int* __restrict__ sidx,
                                  int N, int np, int total,
                                  float* __restrict__ nxyz) {
    int t = blockIdx.x * blockDim.x + threadIdx.x;
    if (t >= total) return;
    const int b = t / (np * 3);
    const int rem = t - b * np * 3;
    const int j = rem / 3, c = rem - j * 3;
    nxyz[t] = xyz[((size_t)b * N + sidx[b * np + j]) * 3 + c];
}

// Ball query: first NSAMP in-range indices in ascending order, pad with first.
__global__ void ballquery_kernel(const float* __restrict__ nxyz,
                                 const float* __restrict__ xyz,
                                 int N, int np, int total, float r2,
                                 int* __restrict__ out) {
    int t = blockIdx.x * blockDim.x + threadIdx.x;
    if (t >= total) return;
    const int b = t / np;
    const float cx = nxyz[t * 3 + 0], cy = nxyz[t * 3 + 1], cz = nxyz[t * 3 + 2];
    const float* px = xyz + (size_t)b * N * 3;
    int* o = out + (size_t)t * NSAMP;
    int cnt = 0, first = -1;
    for (int i = 0; i < N && cnt < NSAMP; ++i) {
        const float dx = px[i * 3 + 0] - cx;
        const float dy = px[i * 3 + 1] - cy;
        const float dz = px[i * 3 + 2] - cz;
        if (dx * dx + dy * dy + dz * dz <= r2) {
            if (first < 0) first = i;
            o[cnt++] = i;
        }
    }
    if (first < 0) first = 0;
    for (; cnt < NSAMP; ++cnt) o[cnt] = first;
}

// ---------------------------------------------------------------------------
// WMMA layer: sOut = relu(sIn * W + bias). sIn is LDS (A-side zero-padded to
// a multiple of 4 columns), W/bias global. SROWS x K times K x Nc.
// Nc multiple of 32 (two output tiles per wave). Guard-free main K loop;
// branch-free clamped tail only when K % 4 != 0 (layer 0 only).
// ---------------------------------------------------------------------------
__device__ __forceinline__ void wmma_layer(const float* sIn, int ldin, int K,
                                           const float* __restrict__ W,
                                           const float* __restrict__ bias, int Nc,
                                           float* sOut, int ldout, int tid) {
    const int wid = tid >> 5, lane = tid & 31, nw = blockDim.x >> 5;
    const int half = lane >> 4, l16 = lane & 15;
    const int mt = SROWS / 16, ntp = Nc / 32, tiles = mt * ntp;
    const int K4 = K & ~3;
    for (int t = wid; t < tiles; t += nw) {
        const int tm = t % mt, tnp = t / mt;
        const int n0 = tnp * 32 + l16;
        const int n1 = n0 + 16;
        // A pair (k0, k0+1) is contiguous & 8B aligned in LDS.
        const float* arow = sIn + (size_t)(tm * 16 + l16) * ldin + 2 * half;
        v8f acc0 = {}, acc1 = {};
#pragma unroll 2
        for (int k = 0; k < K4; k += 4) {
            const v2f a = *(const v2f*)(arow + k);
            const float* wrow = W + (size_t)(k + 2 * half) * Nc;
            v2f b0, b1;
            b0.x = wrow[n0];      b0.y = wrow[Nc + n0];
            b1.x = wrow[n1];      b1.y = wrow[Nc + n1];
            acc0 = wmma_f32_k4(a, b0, acc0);
            acc1 = wmma_f32_k4(a, b1, acc1);
        }
        if (K4 < K) {
            // A side: zero-padded LDS, read directly. B side: clamped address,
            // unconditional load + value select (no EXEC divergence).
            const int k0 = K4 + 2 * half, k1 = k0 + 1;
            const v2f a = *(const v2f*)(arow + K4);
            const int kc0 = (k0 < K) ? k0 : 0;
            const int kc1 = (k1 < K) ? k1 : 0;
            v2f b0, b1;
            float w00 = W[(size_t)kc0 * Nc + n0];
            float w01 = W[(size_t)kc1 * Nc + n0];
            float w10 = W[(size_t)kc0 * Nc + n1];
            float w11 = W[(size_t)kc1 * Nc + n1];
            b0.x = (k0 < K) ? w00 : 0.0f;  b0.y = (k1 < K) ? w01 : 0.0f;
            b1.x = (k0 < K) ? w10 : 0.0f;  b1.y = (k1 < K) ? w11 : 0.0f;
            acc0 = wmma_f32_k4(a, b0, acc0);
            acc1 = wmma_f32_k4(a, b1, acc1);
        }
        const float bb0 = bias[n0];
        const float bb1 = bias[n1];
#pragma unroll
        for (int v = 0; v < 8; ++v) {
            const int m = tm * 16 + v + 8 * half;
            float v0 = acc0[v] + bb0; v0 = v0 > 0.0f ? v0 : 0.0f;
            float v1 = acc1[v] + bb1; v1 = v1 > 0.0f ? v1 : 0.0f;
            sOut[(size_t)m * ldout + n0] = v0;
            sOut[(size_t)m * ldout + n1] = v1;
        }
    }
}

// ---------------------------------------------------------------------------
// Fused SA stage: gather grouped features into LDS (zero-padding ragged cin
// up to a multiple of 4), 3 WMMA MLP layers ping-ponging in LDS, then
// max-pool over NSAMP neighbors.
// grid = (np/TPTS, B), block = SA_THREADS, dynamic LDS = SROWS*(wIn+wOut)*4.
// ---------------------------------------------------------------------------
__global__ void sa_fused_kernel(const float* __restrict__ xyz,
                                const float* __restrict__ prevpts, int prevC,
                                const int* __restrict__ ballidx,
                                const float* __restrict__ nxyz,
                                int N, int np,
                                const float* __restrict__ W0, const float* __restrict__ b0, int c0,
                                const float* __restrict__ W1, const float* __restrict__ b1, int c1,
                                const float* __restrict__ W2, const float* __restrict__ b2, int c2,
                                int wIn, int wOut,
                                float* __restrict__ out) {
    extern __shared__ float smem[];
    float* bufIn  = smem;
    float* bufOut = smem + (size_t)SROWS * wIn;
    const int tid  = threadIdx.x;
    const int b    = blockIdx.y;
    const int base = blockIdx.x * TPTS;
    const int cin  = 3 + prevC;
    const int cinp = (cin + 3) & ~3;   // zero-padded A width for layer 0

    // Gather grouped features (cols [cin, cinp) zero-filled).
    for (int e = tid; e < SROWS * cinp; e += blockDim.x) {
        const int r = e / cinp, c = e - r * cinp;
        const int pt = base + r / NSAMP;
        float v = 0.0f;
        if (c < cin) {
            const int gi = ballidx[((size_t)b * np + pt) * NSAMP + (r % NSAMP)];
            if (c < 3) {
                v = xyz[((size_t)b * N + gi) * 3 + c] - nxyz[((size_t)b * np + pt) * 3 + c];
            } else {
                v = prevpts[((size_t)b * N + gi) * prevC + (c - 3)];
            }
        }
        bufIn[(size_t)r * wIn + c] = v;
    }
    __syncthreads();

    wmma_layer(bufIn,  wIn,  cin, W0, b0, c0, bufOut, wOut, tid);
    __syncthreads();
    wmma_layer(bufOut, wOut, c0,  W1, b1, c1, bufIn,  wIn,  tid);
    __syncthreads();
    wmma_layer(bufIn,  wIn,  c1,  W2, b2, c2, bufOut, wOut, tid);
    __syncthreads();

    // Max-pool over NSAMP rows per sampled point.
    for (int e = tid; e < TPTS * c2; e += blockDim.x) {
        const int pt = e / c2, c = e - pt * c2;
        const float* rbase = bufOut + (size_t)(pt * NSAMP) * wOut + c;
        float m = -3.4e38f;
        for (int s = 0; s < NSAMP; ++s) m = fmaxf(m, rbase[(size_t)s * wOut]);
        out[((size_t)b * np + base + pt) * c2 + c] = m;
    }
}

// ---------------------------------------------------------------------------
// 3-NN interpolation (feature propagation input).
// ---------------------------------------------------------------------------
__global__ void three_interp_kernel(const float* __restrict__ l0xyz,
                                    const float* __restrict__ lxyz,
                                    const float* __restrict__ lpts,
                                    int np, int C, int total,
                                    float* __restrict__ interp) {
    int t = blockIdx.x * blockDim.x + threadIdx.x;
    if (t >= total) return;
    const int b = t / N0;
    const float px = l0xyz[t * 3 + 0], py = l0xyz[t * 3 + 1], pz = l0xyz[t * 3 + 2];
    const float* q = lxyz + (size_t)b * np * 3;
    float d0 = 1e30f, d1 = 1e30f, d2 = 1e30f;
    int   i0 = 0, i1 = 0, i2 = 0;
    for (int i = 0; i < np; ++i) {
        const float dx = q[i * 3 + 0] - px;
        const float dy = q[i * 3 + 1] - py;
        const float dz = q[i * 3 + 2] - pz;
        const float d = dx * dx + dy * dy + dz * dz;
        if (d < d0)      { d2 = d1; i2 = i1; d1 = d0; i1 = i0; d0 = d; i0 = i; }
        else if (d < d1) { d2 = d1; i2 = i1; d1 = d;  i1 = i; }
        else if (d < d2) { d2 = d;  i2 = i; }
    }
    float w0 = 1.0f / (d0 + 1e-8f);
    float w1 = 1.0f / (d1 + 1e-8f);
    float w2 = 1.0f / (d2 + 1e-8f);
    const float ws = w0 + w1 + w2;
    w0 /= ws; w1 /= ws; w2 /= ws;
    const float* p0 = lpts + ((size_t)b * np + i0) * C;
    const float* p1 = lpts + ((size_t)b * np + i1) * C;
    const float* p2 = lpts + ((size_t)b * np + i2) * C;
    float* o = interp + (size_t)t * C;
    for (int c = 0; c < C; ++c) o[c] = w0 * p0[c] + w1 * p1[c] + w2 * p2[c];
}

// ---------------------------------------------------------------------------
// Generic WMMA GEMM: C[:, coloff:coloff+Nc] = act(A(MxK) * W(KxNc) + bias).
// M mult of 16, Nc mult of 32. Guard-free main K loop; clamped tail.
// ---------------------------------------------------------------------------
__global__ void gemm_relu_kernel(const float* __restrict__ A, int M, int K,
                                 const float* __restrict__ W,
                                 const float* __restrict__ bias,
                                 float* __restrict__ C, int Nc, int ldc,
                                 int coloff, int do_relu) {
    const int tid = threadIdx.x;
    const int wid = tid >> 5, lane = tid & 31;
    const int gw = blockIdx.x * (blockDim.x >> 5) + wid;
    const int nw = gridDim.x * (blockDim.x >> 5);
    const int half = lane >> 4, l16 = lane & 15;
    const int mt = M / 16, ntp = Nc / 32, tiles = mt * ntp;
    const int K4 = K & ~3;
    for (int t = gw; t < tiles; t += nw) {
        const int tm = t % mt, tnp = t / mt;
        const int n0 = tnp * 32 + l16;
        const int n1 = n0 + 16;
        const float* arow = A + (size_t)(tm * 16 + l16) * K + 2 * half;
        v8f acc0 = {}, acc1 = {};
#pragma unroll 2
        for (int k = 0; k < K4; k += 4) {
            v2f a;
            a.x = arow[k];
            a.y = arow[k + 1];
            const float* wrow = W + (size_t)(k + 2 * half) * Nc;
            v2f b0, b1;
            b0.x = wrow[n0];      b0.y = wrow[Nc + n0];
            b1.x = wrow[n1];      b1.y = wrow[Nc + n1];
            acc0 = wmma_f32_k4(a, b0, acc0);
            acc1 = wmma_f32_k4(a, b1, acc1);
        }
        if (K4 < K) {
            const int k0 = K4 + 2 * half, k1 = k0 + 1;
            const int kc0 = (k0 < K) ? k0 : 0;
            const int kc1 = (k1 < K) ? k1 : 0;
            const float* ab = A + (size_t)(tm * 16 + l16) * K;
            float a0 = ab[kc0], a1 = ab[kc1];
            v2f a;
            a.x = (k0 < K) ? a0 : 0.0f;
            a.y = (k1 < K) ? a1 : 0.0f;
            v2f b0, b1;
            float w00 = W[(size_t)kc0 * Nc + n0];
            float w01 = W[(size_t)kc1 * Nc + n0];
            float w10 = W[(size_t)kc0 * Nc + n1];
            float w11 = W[(size_t)kc1 * Nc + n1];
            b0.x = (k0 < K) ? w00 : 0.0f;  b0.y = (k1 < K) ? w01 : 0.0f;
            b1.x = (k0 < K) ? w10 : 0.0f;  b1.y = (k1 < K) ? w11 : 0.0f;
            acc0 = wmma_f32_k4(a, b0, acc0);
            acc1 = wmma_f32_k4(a, b1, acc1);
        }
        const float bb0 = bias[n0];
        const float bb1 = bias[n1];
#pragma unroll
        for (int v = 0; v < 8; ++v) {
            const int m = tm * 16 + v + 8 * half;
            float v0 = acc0[v] + bb0;
            float v1 = acc1[v] + bb1;
            if (do_relu) { v0 = v0 > 0.0f ? v0 : 0.0f; v1 = v1 > 0.0f ? v1 : 0.0f; }
            C[(size_t)m * ldc + coloff + n0] = v0;
            C[(size_t)m * ldc + coloff + n1] = v1;
        }
    }
}

// feat = concat(up1(64), up2(64), up3(64), pooled1(64), l0_xyz(3)) -> 259
__global__ void concat_feat_kernel(const float* __restrict__ up1,
                                   const float* __restrict__ up2,
                                   const float* __restrict__ up3,
                                   const float* __restrict__ p1,
                                   const float* __restrict__ pc,
                                   int total, float* __restrict__ feat) {
    int t = blockIdx.x * blockDim.x + threadIdx.x;
    if (t >= total) return;
    const int row = t / 259, c = t - row * 259;
    float v;
    if      (c < 64)  v = up1[(size_t)row * 64 + c];
    else if (c < 128) v = up2[(size_t)row * 64 + (c - 64)];
    else if (c < 192) v = up3[(size_t)row * 64 + (c - 128)];
    else if (c < 256) v = p1 [(size_t)row * 64 + (c - 192)];
    else              v = pc [(size_t)row * 3  + (c - 256)];
    feat[t] = v;
}

// Final FC: 64 -> 3, no relu.
__global__ void fc2_kernel(const float* __restrict__ x,
                           const float* __restrict__ W,
                           const float* __restrict__ b,
                           int rows, float* __restrict__ out) {
    int t = blockIdx.x * blockDim.x + threadIdx.x;
    if (t >= rows * 3) return;
    const int r = t / 3, o = t - r * 3;
    float s = b[o];
    for (int k = 0; k < 64; ++k) s += x[(size_t)r * 64 + k] * W[k * 3 + o];
    out[t] = s;
}

// ---------------------------------------------------------------------------
// Host side
// ---------------------------------------------------------------------------
static inline int cdiv(int a, int b) { return (a + b - 1) / b; }

extern "C" void kernel_launch(void* const* d_in, const int* in_sizes, int n_in,
                              void* d_out, int out_size, void* d_ws, size_t ws_size,
                              hipStream_t stream) {
    (void)in_sizes; (void)n_in; (void)out_size; (void)ws_size;
    const float* pc = (const float*)d_in[0];

    // ---- unpack params (pytree order) ----
    const float *saW[4][3], *saB[4][3];
    int p = 1;
    for (int l = 0; l < 4; ++l)
        for (int j = 0; j < 3; ++j) {
            saW[l][j] = (const float*)d_in[p++];
            saB[l][j] = (const float*)d_in[p++];
        }
    const float *fpW[3], *fpB[3];
    for (int k = 0; k < 3; ++k) {
        fpW[k] = (const float*)d_in[p++];
        fpB[k] = (const float*)d_in[p++];
    }
    const float *upW[4][2], *upB[4][2];
    for (int r = 0; r < 4; ++r)
        for (int j = 0; j < 2; ++j) {
            upW[r][j] = (const float*)d_in[p++];
            upB[r][j] = (const float*)d_in[p++];
        }
    const float* fcW1 = (const float*)d_in[p++];
    const float* fcB1 = (const float*)d_in[p++];
    const float* fcW2 = (const float*)d_in[p++];
    const float* fcB2 = (const float*)d_in[p++];

    // ---- level configs ----
    const int Nout[4] = {4096, 2048, 1024, 512};
    const int cdim[4][3] = {{32,32,64},{64,64,128},{128,128,256},{256,256,512}};
    const int prevC[4] = {0, 64, 128, 256};
    const float radii[4] = {0.05f, 0.10f, 0.20f, 0.30f};

    // ---- workspace bump allocator ----
    char* wp = (char*)d_ws;
    auto alloc = [&](size_t bytes) -> void* {
        void* r = (void*)wp;
        wp += (bytes + 255) & ~(size_t)255;
        return r;
    };
    int*   fpsidx = (int*)alloc((size_t)BB * N0 * sizeof(int));
    int*   ball   = (int*)alloc((size_t)BB * N0 * NSAMP * sizeof(int));
    float* xyzL[4]; float* ptsL[4];
    for (int l = 0; l < 4; ++l) {
        xyzL[l] = (float*)alloc((size_t)BB * Nout[l] * 3 * sizeof(float));
        ptsL[l] = (float*)alloc((size_t)BB * Nout[l] * cdim[l][2] * sizeof(float));
    }
    float* interp  = (float*)alloc((size_t)BB * N0 * 512 * sizeof(float));
    float* upo[3];
    for (int k = 0; k < 3; ++k) upo[k] = (float*)alloc((size_t)BB * N0 * 64 * sizeof(float));
    float* feat    = (float*)alloc((size_t)BB * N0 * 259 * sizeof(float));
    float* binterm = (float*)alloc((size_t)BB * N0 * 256 * sizeof(float));
    float* xcat    = (float*)alloc((size_t)BB * N0 * 512 * sizeof(float));
    float* fc1o    = (float*)alloc((size_t)BB * N0 * 64 * sizeof(float));

    // Allow large dynamic LDS for the fused SA kernel (level 4 ~198 KB;
    // CDNA5 WGP has 320 KB).
    (void)hipFuncSetAttribute(reinterpret_cast<const void*>(sa_fused_kernel),
                              hipFuncAttributeMaxDynamicSharedMemorySize,
                              (int)((size_t)SROWS * (260 + 512) * sizeof(float)));

    // ---- SA levels ----
    const float* curxyz = pc;
    const float* curpts = nullptr;
    int curN = N0;
    for (int l = 0; l < 4; ++l) {
        const int np = Nout[l];
        const int cin = 3 + prevC[l];
        const float r2 = radii[l] * radii[l];

        fps_kernel<<<BB, 256, 0, stream>>>(curxyz, curN, np, fpsidx);

        {
            const int total = BB * np * 3;
            gather_xyz_kernel<<<cdiv(total, 256), 256, 0, stream>>>(
                curxyz, fpsidx, curN, np, total, xyzL[l]);
        }
        {
            const int total = BB * np;
            ballquery_kernel<<<cdiv(total, 256), 256, 0, stream>>>(
                xyzL[l], curxyz, curN, np, total, r2, ball);
        }
        const int c0 = cdim[l][0], c1 = cdim[l][1], c2 = cdim[l][2];
        const int wIn  = (((cin > c1 ? cin : c1) + 3) / 4) * 4;
        const int wOut = (c0 > c2 ? c0 : c2);
        const size_t shmem = (size_t)SROWS * (wIn + wOut) * sizeof(float);
        dim3 grid(np / TPTS, BB);
        sa_fused_kernel<<<grid, SA_THREADS, shmem, stream>>>(
            curxyz, curpts, prevC[l], ball, xyzL[l], curN, np,
            saW[l][0], saB[l][0], c0,
            saW[l][1], saB[l][1], c1,
            saW[l][2], saB[l][2], c2,
            wIn, wOut, ptsL[l]);

        curxyz = xyzL[l];
        curpts = ptsL[l];
        curN = np;
    }

    const int M = BB * N0;
    auto launch_gemm = [&](const float* A, int K, const float* W, const float* bias,
                           float* C, int Nc, int ldc, int coloff, int relu) {
        const int tiles = (M / 16) * (Nc / 32);
        int blocks = cdiv(tiles, GEMM_THREADS / 32);
        if (blocks > 2048) blocks = 2048;
        if (blocks < 1) blocks = 1;
        gemm_relu_kernel<<<blocks, GEMM_THREADS, 0, stream>>>(
            A, M, K, W, bias, C, Nc, ldc, coloff, relu);
    };

    // ---- feature propagation: up1(from lvl4), up2(from lvl3), up3(from lvl2) ----
    for (int k = 0; k < 3; ++k) {
        const int lvl = 3 - k;
        const int C = cdim[lvl][2];
        const int np = Nout[lvl];
        three_interp_kernel<<<cdiv(M, 256), 256, 0, stream>>>(
            pc, xyzL[lvl], ptsL[lvl], np, C, M, interp);
        launch_gemm(interp, C, fpW[k], fpB[k], upo[k], 64, 64, 0, 1);
    }

    // ---- feat concat (259 ch) ----
    {
        const int total = M * 259;
        concat_feat_kernel<<<cdiv(total, 256), 256, 0, stream>>>(
            upo[0], upo[1], upo[2], ptsL[0], pc, total, feat);
    }

    // ---- 4 up branches: 259 -> 256 -> 128, written into xcat columns ----
    for (int r = 0; r < 4; ++r) {
        launch_gemm(feat,    259, upW[r][0], upB[r][0], binterm, 256, 256, 0,       1);
        launch_gemm(binterm, 256, upW[r][1], upB[r][1], xcat,    128, 512, r * 128, 1);
    }

    // ---- FC head: 512 -> 64 (relu) -> 3 ----
    launch_gemm(xcat, 512, fcW1, fcB1, fc1o, 64, 64, 0, 1);
    fc2_kernel<<<cdiv(M * 3, 256), 256, 0, stream>>>(
        fc1o, fcW2, fcB2, M, (float*)d_out);
}